// Detect_72335839199672
// MI455X (gfx1250) — compile-verified
//
#include <hip/hip_runtime.h>
#include <hip/hip_bf16.h>
#include <stdint.h>

#define B_     16
#define P_     16320
#define C_     81
#define NCLS   80          // positive classes
#define TOPK_  400
#define NMSW   13          // ceil(400/32)
#define K1_T   128         // K1 block size = priors per block (B_*P_ % 128 == 0)

// ---------------- CDNA5 async global->LDS staging ----------------
#if defined(__has_builtin)
#  if __has_builtin(__builtin_amdgcn_global_load_async_to_lds_b128)
#    define HAVE_ASYNC_B128 1
#  endif
#  if __has_builtin(__builtin_amdgcn_s_wait_asynccnt)
#    define HAVE_WAIT_ASYNC 1
#  endif
#endif

typedef int v4i_ __attribute__((ext_vector_type(4)));
typedef __attribute__((address_space(1))) v4i_* gv4i_p;   // global v4i*
typedef __attribute__((address_space(3))) v4i_* lv4i_p;   // LDS v4i*

// 16B global -> LDS. Async path uses GLOBAL_LOAD_ASYNC_TO_LDS_B128 (ASYNCcnt).
__device__ __forceinline__ void stage16(void* lds, const void* g) {
#if defined(HAVE_ASYNC_B128)
  __builtin_amdgcn_global_load_async_to_lds_b128((gv4i_p)g, (lv4i_p)lds, 0, 0);
#else
  *(float4*)lds = *(const float4*)g;
#endif
}
__device__ __forceinline__ void wait_stage() {
#if defined(HAVE_ASYNC_B128)
#  if defined(HAVE_WAIT_ASYNC)
  __builtin_amdgcn_s_wait_asynccnt(0);
#  else
  asm volatile("s_wait_asynccnt 0" ::: "memory");
#  endif
#endif
}

// monotonic float<->uint order map (total order, works for negatives)
__device__ __forceinline__ unsigned fmap(float f) {
  unsigned u = __float_as_uint(f);
  return u ^ ((u & 0x80000000u) ? 0xFFFFFFFFu : 0x80000000u);
}
__device__ __forceinline__ float funmap(unsigned u) {
  u ^= (u & 0x80000000u) ? 0x80000000u : 0xFFFFFFFFu;
  return __uint_as_float(u);
}

// ================= K1: fused decode + softmax + mask =================
// odm_conf is tiled through LDS: 128 consecutive priors' rows form one
// contiguous 41,472 B span -> coalesced async copy, softmax runs from LDS
// (word t*81+c, gcd(81,64)=1 -> bank-conflict-free).
__global__ void __launch_bounds__(K1_T)
k_decode(const float* __restrict__ arm_loc,
         const float* __restrict__ arm_conf,
         const float* __restrict__ odm_loc,
         const float* __restrict__ odm_conf,
         const float* __restrict__ priors,
         float* __restrict__ boxes,      // (B,P,4) corner form
         float* __restrict__ scoresT) {  // (B,80,P) masked
  __shared__ __align__(16) float tile[K1_T * C_];   // 41,472 B

  int tid = threadIdx.x;
  int t0  = blockIdx.x * K1_T;
  int t   = t0 + tid;                    // flat (b,p); grid covers exactly B_*P_
  int b = t / P_, p = t % P_;

  // stage this block's odm_conf rows (contiguous span) into LDS
  const float* src = odm_conf + (size_t)t0 * C_;
  for (int c = tid; c < (K1_T * C_) / 4; c += K1_T)
    stage16(&tile[c * 4], src + c * 4);
  wait_stage();
  __syncthreads();

  // ARM positive flag: softmax[1] > 0.01  <=>  a1 - a0 > -ln(99)
  float a0 = arm_conf[(size_t)t * 2 + 0];
  float a1 = arm_conf[(size_t)t * 2 + 1];
  bool flag = (a1 - a0) > -4.5951198501345898f;

  // center_size(decode(arm)) collapses to direct center-size refine
  float4 pr = *(const float4*)(priors + (size_t)p * 4);
  float4 al = *(const float4*)(arm_loc + (size_t)t * 4);
  float rcx = pr.x + al.x * 0.1f * pr.z;
  float rcy = pr.y + al.y * 0.1f * pr.w;
  float rw  = pr.z * __expf(al.z * 0.2f);
  float rh  = pr.w * __expf(al.w * 0.2f);

  float4 ol = *(const float4*)(odm_loc + (size_t)t * 4);
  float bcx = rcx + ol.x * 0.1f * rw;
  float bcy = rcy + ol.y * 0.1f * rh;
  float bw  = rw * __expf(ol.z * 0.2f);
  float bh  = rh * __expf(ol.w * 0.2f);
  float4 bx = {bcx - bw * 0.5f, bcy - bh * 0.5f, bcx + bw * 0.5f, bcy + bh * 0.5f};
  *(float4*)(boxes + (size_t)t * 4) = bx;

  // 81-way softmax out of LDS (3 cheap passes, no register blowup)
  const float* oc = &tile[tid * C_];
  float m = -3.0e38f;
  for (int c = 0; c < C_; ++c) m = fmaxf(m, oc[c]);
  float s = 0.f;
  for (int c = 0; c < C_; ++c) s += __expf(oc[c] - m);
  float inv = 1.0f / s;
  // masked score, class-major layout (coalesced across p)
  for (int c = 1; c < C_; ++c) {
    float sc = __expf(oc[c] - m) * inv;
    float ms = (flag && sc > 0.01f) ? sc : -1.0f;
    scoresT[((size_t)(b * NCLS + (c - 1))) * P_ + p] = ms;
  }
}

// ================= K2: per-(image,class) top-400 =================
// LDS-resident radix select (4x8-bit MSB-first) + compaction + bitonic sort
// on 64-bit (~u, idx) keys -> exact jax top_k ordering, deterministic.
__global__ void __launch_bounds__(256)
k_topk(const float* __restrict__ scoresT,
       const float* __restrict__ boxes,
       float* __restrict__ out) {
  extern __shared__ unsigned su[];             // P_ mapped scores (63.75 KB)
  __shared__ unsigned hist[256];
  __shared__ unsigned sel_bin, sel_k;
  __shared__ unsigned cnt_gt, cnt_eq;
  __shared__ unsigned long long key[512];      // (~u)<<32 | idx

  int bc  = blockIdx.x;                        // 0..B*80-1
  int b   = bc / NCLS, cls = bc % NCLS;
  int tid = threadIdx.x;
  const float* sc = scoresT + (size_t)bc * P_;

  // ---- stage whole score vector into LDS (async b128) ----
  for (int c4 = tid; c4 < P_ / 4; c4 += blockDim.x)
    stage16(&su[c4 * 4], sc + c4 * 4);
  wait_stage();
  __syncthreads();
  // map to order-preserving uints in place
  for (int i = tid; i < P_; i += blockDim.x)
    su[i] = fmap(__uint_as_float(su[i]));
  __syncthreads();

  // ---- radix select: value of 400th largest ----
  unsigned prefix = 0, pmask = 0;
  int k = TOPK_;
  for (int shift = 24; shift >= 0; shift -= 8) {
    hist[tid] = 0;
    __syncthreads();
    for (int i = tid; i < P_; i += blockDim.x) {
      unsigned u = su[i];
      if ((u & pmask) == prefix) atomicAdd(&hist[(u >> shift) & 255u], 1u);
    }
    __syncthreads();
    if (tid == 0) {
      int cum = 0, bsel = 0, kk = k;
      for (int bin = 255; bin >= 0; --bin) {
        int h = (int)hist[bin];
        if (cum + h >= kk) { bsel = bin; kk = kk - cum; break; }
        cum += h;
      }
      sel_bin = (unsigned)bsel; sel_k = (unsigned)kk;
    }
    __syncthreads();
    prefix |= sel_bin << shift;
    pmask  |= 0xFFu << shift;
    k = (int)sel_k;
    __syncthreads();
  }
  const unsigned T = prefix;                   // u-value of the 400th largest

  // ---- compaction (order canonicalized by sort below) ----
  if (tid == 0) { cnt_gt = 0; cnt_eq = 0; }
  for (int i = tid; i < 512; i += blockDim.x) key[i] = ~0ull;  // pad -> sorts last
  __syncthreads();
  for (int i = tid; i < P_; i += blockDim.x) {
    unsigned u = su[i];
    if (u > T) {
      unsigned pos = atomicAdd(&cnt_gt, 1u);   // pos < 400 guaranteed
      key[pos] = ((unsigned long long)(~u) << 32) | (unsigned)i;
    }
  }
  __syncthreads();
  unsigned ngt = cnt_gt;                       // = 400 - (#equals to take)
  for (int i = tid; i < P_; i += blockDim.x) {
    unsigned u = su[i];
    if (u == T) {
      unsigned pos = atomicAdd(&cnt_eq, 1u);
      if (ngt + pos < TOPK_)
        key[ngt + pos] = ((unsigned long long)(~u) << 32) | (unsigned)i;
    }
  }
  __syncthreads();

  // ---- bitonic sort ascending on 512 keys (=> score desc, idx asc) ----
  for (unsigned ksz = 2; ksz <= 512; ksz <<= 1) {
    for (unsigned j = ksz >> 1; j > 0; j >>= 1) {
      for (int i = tid; i < 512; i += blockDim.x) {
        unsigned ixj = (unsigned)i ^ j;
        if (ixj > (unsigned)i) {
          bool up = ((i & ksz) == 0);
          unsigned long long a = key[i], c2 = key[ixj];
          if ((a > c2) == up) { key[i] = c2; key[ixj] = a; }
        }
      }
      __syncthreads();
    }
  }

  // ---- emit rows (score, x1,y1,x2,y2) straight into d_out ----
  float* gout = out + (((size_t)b * C_ + (cls + 1)) * TOPK_) * 5;
  for (int j = tid; j < TOPK_; j += blockDim.x) {
    unsigned long long kk = key[j];
    unsigned u  = ~(unsigned)(kk >> 32);
    int idx     = (int)(kk & 0xFFFFFFFFu);
    float v     = funmap(u);
    float4 bx   = *(const float4*)(boxes + ((size_t)b * P_ + idx) * 4);
    float* row  = gout + (size_t)j * 5;
    row[0] = v; row[1] = bx.x; row[2] = bx.y; row[3] = bx.z; row[4] = bx.w;
  }
}

// ================= K3: greedy NMS + stable compaction (in place) =================
__global__ void __launch_bounds__(256)
k_nms(float* __restrict__ out) {
  __shared__ __align__(16) float rows[TOPK_ * 5];   // staged candidate rows
  __shared__ float area[TOPK_];
  __shared__ unsigned masks[TOPK_ * NMSW];          // suppression bitmask (j>i)
  __shared__ unsigned keepw[NMSW];
  __shared__ int posArr[TOPK_];
  __shared__ int nkept;

  int bc  = blockIdx.x;
  int b   = bc / NCLS, cls = bc % NCLS;
  int tid = threadIdx.x;
  float* grows = out + (((size_t)b * C_ + (cls + 1)) * TOPK_) * 5;

  // stage 400 rows (8000 B = 500 x b128) into LDS via async path
  for (int c = tid; c < (TOPK_ * 5) / 4; c += blockDim.x)
    stage16(&rows[c * 4], grows + c * 4);
  wait_stage();
  __syncthreads();

  // areas + candidate bits (score > 0)
  if (tid < NMSW) keepw[tid] = 0;
  __syncthreads();
  for (int i = tid; i < TOPK_; i += blockDim.x) {
    float x1 = rows[i * 5 + 1], y1 = rows[i * 5 + 2];
    float x2 = rows[i * 5 + 3], y2 = rows[i * 5 + 4];
    area[i] = (x2 - x1) * (y2 - y1);
    if (rows[i * 5] > 0.0f) atomicOr(&keepw[i >> 5], 1u << (i & 31));
  }
  __syncthreads();

  // pairwise IoU -> 400 x 13-word suppression bitmasks (balanced over 5200 tasks)
  for (int task = tid; task < TOPK_ * NMSW; task += blockDim.x) {
    int i = task / NMSW, w = task % NMSW;
    float x1i = rows[i * 5 + 1], y1i = rows[i * 5 + 2];
    float x2i = rows[i * 5 + 3], y2i = rows[i * 5 + 4];
    float ai  = area[i];
    unsigned bits = 0;
    int j0 = w * 32;
#pragma unroll
    for (int t2 = 0; t2 < 32; ++t2) {
      int j = j0 + t2;
      if (j < TOPK_ && j > i) {
        float iw = fminf(x2i, rows[j * 5 + 3]) - fmaxf(x1i, rows[j * 5 + 1]);
        float ih = fminf(y2i, rows[j * 5 + 4]) - fmaxf(y1i, rows[j * 5 + 2]);
        iw = fmaxf(iw, 0.f); ih = fmaxf(ih, 0.f);
        float inter = iw * ih;
        float iou = inter / (ai + area[j] - inter);
        if (iou > 0.45f) bits |= (1u << t2);
      }
    }
    masks[i * NMSW + w] = bits;
  }
  __syncthreads();

  // sequential greedy scan, one wave32 in lockstep (13 lanes AND per step)
  if (tid < 32) {
    volatile unsigned* vk = keepw;
    for (int i = 0; i < TOPK_; ++i) {
      unsigned kb = vk[i >> 5];
      if (kb & (1u << (i & 31))) {
        if (tid < NMSW) vk[tid] &= ~masks[i * NMSW + tid];
      }
    }
  }
  __syncthreads();

  // stable positions of kept rows (matches argsort(where(keep,0,1), stable))
  if (tid == 0) {
    int cnt = 0;
    for (int i = 0; i < TOPK_; ++i) {
      bool kp = (keepw[i >> 5] >> (i & 31)) & 1u;
      posArr[i] = kp ? cnt : -1;
      cnt += kp ? 1 : 0;
    }
    nkept = cnt;
  }
  __syncthreads();

  // write kept rows compacted to front, zero the tail (disjoint ranges)
  for (int i = tid; i < TOPK_; i += blockDim.x) {
    int p = posArr[i];
    if (p >= 0) {
      float* dst = grows + (size_t)p * 5;
      dst[0] = rows[i * 5 + 0]; dst[1] = rows[i * 5 + 1];
      dst[2] = rows[i * 5 + 2]; dst[3] = rows[i * 5 + 3];
      dst[4] = rows[i * 5 + 4];
    }
  }
  int nk = nkept;
  for (int j = nk + tid; j < TOPK_; j += blockDim.x) {
    float* dst = grows + (size_t)j * 5;
    dst[0] = 0.f; dst[1] = 0.f; dst[2] = 0.f; dst[3] = 0.f; dst[4] = 0.f;
  }
}

// ================= K4: zero background class =================
__global__ void k_bg(float* __restrict__ out) {
  int t = blockIdx.x * blockDim.x + threadIdx.x;
  const int per = TOPK_ * 5;                  // 2000
  if (t < B_ * per) {
    int b = t / per, r = t % per;
    out[(size_t)b * C_ * per + r] = 0.f;      // class 0 rows
  }
}

// ================= host launcher =================
extern "C" void kernel_launch(void* const* d_in, const int* in_sizes, int n_in,
                              void* d_out, int out_size, void* d_ws, size_t ws_size,
                              hipStream_t stream) {
  (void)in_sizes; (void)n_in; (void)out_size; (void)ws_size;
  const float* arm_loc  = (const float*)d_in[0];
  const float* arm_conf = (const float*)d_in[1];
  const float* odm_loc  = (const float*)d_in[2];
  const float* odm_conf = (const float*)d_in[3];
  const float* priors   = (const float*)d_in[4];
  float* out = (float*)d_out;

  // workspace: boxes (B*P*4 f32 = 4.18 MB) + masked scoresT (B*80*P f32 = 83.6 MB)
  float* boxes   = (float*)d_ws;
  float* scoresT = boxes + (size_t)B_ * P_ * 4;

  int n1 = B_ * P_;                           // 261120 = 2040 * 128
  k_decode<<<n1 / K1_T, K1_T, 0, stream>>>(arm_loc, arm_conf, odm_loc,
                                           odm_conf, priors, boxes, scoresT);
  k_topk<<<B_ * NCLS, 256, P_ * sizeof(unsigned), stream>>>(scoresT, boxes, out);
  k_nms<<<B_ * NCLS, 256, 0, stream>>>(out);
  k_bg<<<(B_ * TOPK_ * 5 + 255) / 256, 256, 0, stream>>>(out);
}